// TableDistance_5669356831001
// MI455X (gfx1250) — compile-verified
//
#include <hip/hip_runtime.h>
#include <hip/hip_bf16.h>

typedef __attribute__((ext_vector_type(2))) float v2f;
typedef __attribute__((ext_vector_type(8))) float v8f;

// Problem constants from the reference: B=4, N=2048, D=64, K=256
#define DD 64
#define KK 256
#define LDS_STRIDE 66   // even -> 8B-aligned float2 LDS loads; (k*66+d)%64 spreads banks

// Block: 128 threads = 4 wave32 waves. Each wave computes 16 rows x 256 cols.
// Block covers 64 rows; grid.x = 8192/64 = 128 blocks for WGP/HBM-channel spread.
__global__ __launch_bounds__(128)
void TableDistance_5669356831001_kernel(const float* __restrict__ x,
                                        const float* __restrict__ table,
                                        const float* __restrict__ scale,
                                        float* __restrict__ out) {
    __shared__ float ldsT[KK * LDS_STRIDE]; // table transposed: ldsT[col*66 + d]
    __shared__ float ldsT2[KK];             // per-column sum of squares

    const int tid = threadIdx.x;

    // ---- Stage table into LDS transposed, computing column norms once ----
    // 128 threads, 256 columns -> each thread owns 2 columns.
    #pragma unroll
    for (int cc = 0; cc < 2; ++cc) {
        const int col = tid + cc * 128;
        float t2 = 0.0f;
        #pragma unroll 4
        for (int d = 0; d < DD; ++d) {
            const float v = table[d * KK + col];   // coalesced along col
            ldsT[col * LDS_STRIDE + d] = v;
            t2 += v * v;
        }
        ldsT2[col] = t2;
    }
    __syncthreads();

    const int wave = tid >> 5;        // 0..3
    const int lane = tid & 31;        // wave32 lane
    const int half = lane >> 4;       // 0: K=0,1  1: K=2,3 (A-matrix 16x4 f32 layout)
    const int m    = lane & 15;       // row-in-tile (A), col-in-tile (B/C/D)

    const int rowBase = blockIdx.x * 64 + wave * 16;

    // ---- Preload A fragments for this wave's 16 rows, D=64 -> 16 K-steps --
    //   lanes 0-15 : a = { x[row, 4s+0], x[row, 4s+1] }
    //   lanes 16-31: a = { x[row, 4s+2], x[row, 4s+3] }
    const float* xrow = x + (size_t)(rowBase + m) * DD + half * 2;
    v2f afrag[16];
    float x2p = 0.0f; // this lane's half of the row sum-of-squares
    #pragma unroll
    for (int s = 0; s < 16; ++s) {
        afrag[s] = *(const v2f*)(xrow + 4 * s);
        x2p += afrag[s].x * afrag[s].x + afrag[s].y * afrag[s].y;
    }
    // lane m and lane m+16 both end up with full ||x_row||^2
    const float x2full = x2p + __shfl_xor(x2p, 16, 32);

    // Redistribute to C/D layout: result VGPR r holds M = r + 8*half.
    float x2m[8];
    #pragma unroll
    for (int r = 0; r < 8; ++r) {
        x2m[r] = __shfl(x2full, r + 8 * half, 32);
    }

    const float sabs = fabsf(scale[0]);

    // ---- Column-tile loop: 16 tiles of 16 cols each -----------------------
    #pragma unroll 1
    for (int c = 0; c < 16; ++c) {
        const int col = c * 16 + m;
        const float* bcol = &ldsT[col * LDS_STRIDE + half * 2];

        v8f acc = {};
        #pragma unroll
        for (int s = 0; s < 16; ++s) {
            const v2f b = *(const v2f*)(bcol + 4 * s);   // ds_load_b64
            acc = __builtin_amdgcn_wmma_f32_16x16x4_f32(
                /*neg_a=*/false, afrag[s],
                /*neg_b=*/false, b,
                /*c_mod=*/(short)0, acc,
                /*reuse_a=*/false, /*reuse_b=*/false);
        }

        // Column norm precomputed during staging; N = lane%16 matches C/D layout,
        // lanes 16-31 broadcast-read the same 16 addresses (no bank conflict).
        const float t2 = ldsT2[col];

        // out = |s| * (x2[M] + t2[N] - 2*dot)
        float* op = out + (size_t)(rowBase + 8 * half) * KK + col;
        #pragma unroll
        for (int r = 0; r < 8; ++r) {
            op[r * KK] = sabs * (x2m[r] + t2 - 2.0f * acc[r]);
        }
    }
}

extern "C" void kernel_launch(void* const* d_in, const int* in_sizes, int n_in,
                              void* d_out, int out_size, void* d_ws, size_t ws_size,
                              hipStream_t stream) {
    const float* x     = (const float*)d_in[0];
    const float* table = (const float*)d_in[1];
    const float* scale = (const float*)d_in[2];
    float* out = (float*)d_out;

    const int rows = in_sizes[0] / DD;   // B*N = 8192
    const int grid = rows / 64;          // 128 blocks of 4 waves

    TableDistance_5669356831001_kernel<<<grid, 128, 0, stream>>>(x, table, scale, out);
}